// SphericalCheb_75883482185958
// MI455X (gfx1250) — compile-verified
//
#include <hip/hip_runtime.h>

typedef __attribute__((ext_vector_type(2))) float v2f;
typedef __attribute__((ext_vector_type(8))) float v8f;

#define FIN 64
#define KCH 5
#define KF  (KCH * FIN)   // 320, feat row stride

// ===========================================================================
// CSR construction (per call; ~3M cheap int atomics instead of 400M f32 ones)
// ===========================================================================

__global__ void zero_i32_kernel(int* __restrict__ a, long long n) {
  long long t = (long long)blockIdx.x * blockDim.x + threadIdx.x;
  if (t < n) a[t] = 0;
}

__global__ void count_kernel(const int* __restrict__ rows, int* __restrict__ cnt,
                             long long E) {
  long long t = (long long)blockIdx.x * blockDim.x + threadIdx.x;
  if (t < E) atomicAdd(&cnt[rows[t]], 1);
}

// per-block inclusive scan in LDS -> exclusive rowptr + block sums
__global__ void scan_block_kernel(const int* __restrict__ cnt, int* __restrict__ rowptr,
                                  int* __restrict__ bsum, long long V) {
  __shared__ int sh[256];
  long long gid = (long long)blockIdx.x * 256 + threadIdx.x;
  int v = (gid < V) ? cnt[gid] : 0;
  sh[threadIdx.x] = v;
  __syncthreads();
  for (int off = 1; off < 256; off <<= 1) {
    int t = (threadIdx.x >= (unsigned)off) ? sh[threadIdx.x - off] : 0;
    __syncthreads();
    sh[threadIdx.x] += t;
    __syncthreads();
  }
  int excl = (threadIdx.x == 0) ? 0 : sh[threadIdx.x - 1];
  if (gid < V) rowptr[gid] = excl;
  if (threadIdx.x == 255) bsum[blockIdx.x] = sh[255];
}

// single-block scan of block sums (nb <= 1024)
__global__ void scan_top_kernel(int* __restrict__ bsum, int nb) {
  __shared__ int sh[1024];
  int tid = threadIdx.x;
  int v = (tid < nb) ? bsum[tid] : 0;
  sh[tid] = v;
  __syncthreads();
  for (int off = 1; off < 1024; off <<= 1) {
    int t = (tid >= off) ? sh[tid - off] : 0;
    __syncthreads();
    sh[tid] += t;
    __syncthreads();
  }
  if (tid < nb) bsum[tid] = (tid == 0) ? 0 : sh[tid - 1];
}

__global__ void add_offsets_kernel(int* __restrict__ rowptr, const int* __restrict__ bsum,
                                   long long V, int Etot) {
  long long gid = (long long)blockIdx.x * 256 + threadIdx.x;
  if (gid < V) rowptr[gid] += bsum[blockIdx.x];
  if (gid == 0) rowptr[V] = Etot;   // sentinel
}

// scatter packed (col, val) pairs into CSR order
__global__ void scatter_kernel(const int* __restrict__ rows, const int* __restrict__ cols,
                               const float* __restrict__ vals,
                               const int* __restrict__ rowptr, int* __restrict__ cursor,
                               int2* __restrict__ csr_pair, long long E) {
  long long t = (long long)blockIdx.x * blockDim.x + threadIdx.x;
  if (t >= E) return;
  int r = rows[t];
  int p = rowptr[r] + atomicAdd(&cursor[r], 1);
  int2 pr;
  pr.x = cols[t];
  pr.y = __float_as_int(vals[t]);
  csr_pair[p] = pr;
}

// ===========================================================================
// Chebyshev pipeline
// ===========================================================================

// feat[:, 0:64] = x   (T0)
__global__ void copy_x_kernel(float* __restrict__ feat, const float* __restrict__ x,
                              long long n16) {
  long long t = (long long)blockIdx.x * blockDim.x + threadIdx.x;
  if (t >= n16) return;
  long long v = t >> 4;
  int q = (int)(t & 15) * 4;
  *(float4*)(feat + v * KF + q) = *(const float4*)(x + v * FIN + q);
}

// Gather SpMM: dst[v,f] = coef * sum_j val_j * src[col_j, f]  - prev[v,f]
// 16 threads per vertex, float4 per thread, zero atomics. Gather source is
// a 48MB block -> L2-resident (192MB L2).
__global__ void cheb_spmm_csr_kernel(const int* __restrict__ rowptr,
                                     const int2* __restrict__ csr_pair,
                                     const float* __restrict__ src,   // feat+(k-1)*64
                                     const float* __restrict__ prev,  // feat+(k-2)*64 or null
                                     float* __restrict__ dst,         // feat+k*64
                                     float coef, long long n16) {
  long long t = (long long)blockIdx.x * blockDim.x + threadIdx.x;
  if (t >= n16) return;
  long long v = t >> 4;
  int q = (int)(t & 15) * 4;
  int beg = rowptr[v];
  int end = rowptr[v + 1];
  float ax = 0.f, ay = 0.f, az = 0.f, aw = 0.f;
  for (int j = beg; j < end; ++j) {
    int2 p = csr_pair[j];               // one b64: (col, val)
    float s = __int_as_float(p.y);
    const float4 xv = *(const float4*)(src + (long long)p.x * KF + q);
    ax = fmaf(s, xv.x, ax);
    ay = fmaf(s, xv.y, ay);
    az = fmaf(s, xv.z, az);
    aw = fmaf(s, xv.w, aw);
  }
  float4 o;
  if (prev) {
    const float4 pv = *(const float4*)(prev + v * KF + q);
    o = make_float4(coef * ax - pv.x, coef * ay - pv.y,
                    coef * az - pv.z, coef * aw - pv.w);
  } else {
    o = make_float4(coef * ax, coef * ay, coef * az, coef * aw);
  }
  *(float4*)(dst + v * KF + q) = o;
}

// ===========================================================================
// Dense combine + PReLU with f32 WMMA (V_WMMA_F32_16X16X4_F32), unchanged.
// ===========================================================================
__global__ void cheb_gemm_prelu_kernel(const float* __restrict__ feat,
                                       const float* __restrict__ W,   // [320][64]
                                       const float* __restrict__ alpha,
                                       float* __restrict__ out) {
  const int m_base = blockIdx.x * 16;
  const int wave   = threadIdx.x >> 5;   // 0..3
  const int lane   = threadIdx.x & 31;
  const int n0     = wave * 16;
  const int lrow   = lane & 15;
  const int khalf  = lane >> 4;          // 0 or 1

  const float* arow = feat + (long long)(m_base + lrow) * KF + 2 * khalf;
  const float* bcol = W + (long long)(2 * khalf) * FIN + n0 + lrow;

  v8f c = {};
#pragma unroll 4
  for (int k = 0; k < KF; k += 4) {
    v2f a = *(const v2f*)(arow + k);
    v2f b;
    b.x = bcol[k * FIN];
    b.y = bcol[k * FIN + FIN];
    c = __builtin_amdgcn_wmma_f32_16x16x4_f32(
        /*neg_a=*/false, a, /*neg_b=*/false, b,
        /*c_mod=*/(short)0, c, /*reuse_a=*/false, /*reuse_b=*/false);
  }

  const float al = alpha[0];
  float* orow = out + (long long)(m_base + 8 * khalf) * FIN + n0 + lrow;
#pragma unroll
  for (int j = 0; j < 8; ++j) {
    float v = c[j];
    v = v > 0.f ? v : al * v;
    orow[(long long)j * FIN] = v;
  }
}

// ===========================================================================
extern "C" void kernel_launch(void* const* d_in, const int* in_sizes, int n_in,
                              void* d_out, int out_size, void* d_ws, size_t ws_size,
                              hipStream_t stream) {
  const float* x     = (const float*)d_in[0];
  const float* lap   = (const float*)d_in[1];
  const float* W     = (const float*)d_in[2];
  const float* alpha = (const float*)d_in[3];
  const int*   rows  = (const int*)d_in[4];
  const int*   cols  = (const int*)d_in[5];

  const long long E = in_sizes[1];
  const long long V = (long long)in_sizes[0] / FIN;

  // workspace layout (256B-aligned chunks)
  char* ws = (char*)d_ws;
  auto take = [&](size_t bytes) {
    char* p = ws;
    ws += (bytes + 255) & ~(size_t)255;
    return p;
  };
  float* feat     = (float*)take((size_t)V * KF * sizeof(float));   // [V][320]
  int*   rowptr   = (int*)  take((size_t)(V + 1) * sizeof(int));
  int*   cnt      = (int*)  take((size_t)V * 2 * sizeof(int));      // cnt + cursor
  int*   cursor   = cnt + V;
  int*   bsum     = (int*)  take(1024 * sizeof(int));
  int2*  csr_pair = (int2*) take((size_t)E * sizeof(int2));

  float* out = (float*)d_out;

  const int tb = 256;
  const long long nb = (V + 255) / 256;               // scan blocks (<=1024)
  dim3 gV((unsigned)nb);
  dim3 gE((unsigned)((E + tb - 1) / tb));
  dim3 g2V((unsigned)((2 * V + tb - 1) / tb));

  // ---- CSR build ----
  zero_i32_kernel<<<g2V, tb, 0, stream>>>(cnt, 2 * V);               // cnt + cursor
  count_kernel<<<gE, tb, 0, stream>>>(rows, cnt, E);
  scan_block_kernel<<<gV, tb, 0, stream>>>(cnt, rowptr, bsum, V);
  scan_top_kernel<<<1, 1024, 0, stream>>>(bsum, (int)nb);
  add_offsets_kernel<<<gV, tb, 0, stream>>>(rowptr, bsum, V, (int)E);
  scatter_kernel<<<gE, tb, 0, stream>>>(rows, cols, lap, rowptr, cursor, csr_pair, E);

  // ---- Chebyshev recurrence (gather form, no atomics) ----
  const long long n16 = V * 16;
  dim3 gF((unsigned)((n16 + tb - 1) / tb));
  copy_x_kernel<<<gF, tb, 0, stream>>>(feat, x, n16);
  for (int k = 1; k < KCH; ++k) {
    const float* src  = feat + (long long)(k - 1) * FIN;
    const float* prev = (k >= 2) ? feat + (long long)(k - 2) * FIN : nullptr;
    float*       dst  = feat + (long long)k * FIN;
    cheb_spmm_csr_kernel<<<gF, tb, 0, stream>>>(rowptr, csr_pair, src, prev, dst,
                                                (k == 1) ? 1.f : 2.f, n16);
  }

  // ---- out = PReLU(feat @ W) via f32 WMMA ----
  dim3 gg((unsigned)(V / 16));
  cheb_gemm_prelu_kernel<<<gg, 128, 0, stream>>>(feat, W, alpha, out);
}